// GCN_NodeClassification_75479755259980
// MI455X (gfx1250) — compile-verified
//
#include <hip/hip_runtime.h>

typedef __attribute__((ext_vector_type(2))) float v2f;
typedef __attribute__((ext_vector_type(8))) float v8f;

#define BN_EPS 1e-5f
#define NEG_SLOPE 0.01f

// ---------------------------------------------------------------- utilities
__global__ void zero_f32(float* __restrict__ p, int n) {
  int i = blockIdx.x * blockDim.x + threadIdx.x;
  if (i < n) p[i] = 0.0f;
}

// degree accumulation: deg[dst[e]] += 1
__global__ void deg_accum(const long long* __restrict__ ei, float* __restrict__ deg, int E) {
  int e = blockIdx.x * blockDim.x + threadIdx.x;
  if (e < E) atomicAdd(&deg[(int)ei[(long long)E + e]], 1.0f);
}

// dis[i] = rsqrt(deg[i] + 1)  (in place over the deg buffer)
__global__ void deg_to_dis(float* __restrict__ dis, int n) {
  int i = blockIdx.x * blockDim.x + threadIdx.x;
  if (i < n) dis[i] = rsqrtf(dis[i] + 1.0f);
}

// ------------------------------------------------- WMMA fp32 GEMM: H = X * W
// X: [N,128] row-major, W: [128,NO] row-major, H: [N,NO] (no bias).
// Block = 256 threads (8 waves). Block covers 128 rows; each wave owns a
// 16-row x NO-col strip = NO/16 accumulator tiles, K stepped by 4 with
// v_wmma_f32_16x16x4_f32. W is staged transposed in LDS so the B fragment
// (k, k+1 at fixed n) is one 8-byte ds load.
// Out-of-range rows are CLAMPED (not branched): row M of C depends only on
// row M of A, and the epilogue guards stores with r < N, so clamped-duplicate
// rows are harmless. This keeps the K-loop branch-free with EXEC all-1s.
template <int NO>
__launch_bounds__(256)
__global__ void gemm_wmma_f32(const float* __restrict__ X, const float* __restrict__ W,
                              float* __restrict__ H, int N) {
  constexpr int K = 128;
  constexpr int NT = NO / 16;
  __shared__ float Wt[NO * K];  // Wt[n*K + k] = W[k*NO + n]

  const int tid = threadIdx.x;
  for (int idx = tid; idx < K * NO; idx += 256) {
    int k = idx / NO, n = idx % NO;
    Wt[n * K + k] = W[idx];
  }
  __syncthreads();

  const int wave = tid >> 5;
  const int lane = tid & 31;
  const int l15  = lane & 15;
  const int koff = (lane >> 4) << 1;          // 0 for lanes 0-15, 2 for 16-31
  const int rowBase = blockIdx.x * 128 + wave * 16;
  int arow = rowBase + l15;                   // A-fragment source row
  if (arow > N - 1) arow = N - 1;             // clamp: branch-free hot loop
  const float* __restrict__ xrow = X + (long long)arow * K + koff;

  v8f acc[NT] = {};

  for (int k0 = 0; k0 < K; k0 += 4) {
    v2f a = *(const v2f*)(xrow + k0);          // K = k0+koff, k0+koff+1
#pragma unroll
    for (int t = 0; t < NT; ++t) {
      v2f b = *(const v2f*)(&Wt[(t * 16 + l15) * K + k0 + koff]);
      acc[t] = __builtin_amdgcn_wmma_f32_16x16x4_f32(
          /*neg_a=*/false, a, /*neg_b=*/false, b,
          /*c_mod=*/(short)0, acc[t], /*reuse_a=*/false, /*reuse_b=*/false);
    }
  }

  // C/D layout: VGPR v -> M = v + 8*(lane>=16); N = lane&15
  const int mofs = (lane >> 4) * 8;
#pragma unroll
  for (int t = 0; t < NT; ++t) {
#pragma unroll
    for (int v = 0; v < 8; ++v) {
      int r = rowBase + mofs + v;
      if (r < N) H[(long long)r * NO + t * 16 + l15] = acc[t][v];
    }
  }
}

// ------------------------------------------- self-loop + bias initialization
// out[i,f] = h[i,f] * dis[i]^2 + bias[f]   (float4-vectorized, F % 4 == 0)
__global__ void agg_init_v4(const float4* __restrict__ h4, const float* __restrict__ dis,
                            const float4* __restrict__ bias4, float4* __restrict__ out4,
                            int N, int F4) {
  int idx = blockIdx.x * blockDim.x + threadIdx.x;
  if (idx >= N * F4) return;
  int i = idx / F4, f = idx % F4;
  float d = dis[i];
  float d2 = d * d;
  float4 hv = h4[idx];
  float4 bv = bias4[f];
  float4 o;
  o.x = hv.x * d2 + bv.x;
  o.y = hv.y * d2 + bv.y;
  o.z = hv.z * d2 + bv.z;
  o.w = hv.w * d2 + bv.w;
  out4[idx] = o;
}

// ------------------------------------------------- edge gather/scatter-add
// One wave per edge; lane j handles features {j, j+32, ...} -> coalesced
// 128B gather of h[src] and stride-4B global_atomic_add_f32 into out[dst]
// (2 cache lines per atomic instruction).
template <int F>
__global__ void edge_scatter(const long long* __restrict__ ei, const float* __restrict__ dis,
                             const float* __restrict__ h, float* __restrict__ out, int E) {
  int gid = blockIdx.x * blockDim.x + threadIdx.x;
  int e = gid >> 5;
  int lane = gid & 31;
  if (e >= E) return;
  int s = (int)ei[e];
  int d = (int)ei[(long long)E + e];
  float nrm = dis[s] * dis[d];
  const float* __restrict__ hs = h + (long long)s * F;
  float* __restrict__ od = out + (long long)d * F;
#pragma unroll
  for (int j = 0; j < F / 32; ++j) {
    int f = lane + j * 32;
    atomicAdd(&od[f], hs[f] * nrm);
  }
}

// ------------------------------------------------------------ batch norm
// stats[0:128]=col sums, stats[128:256]=col sum of squares,
// stats[256:384]=scale, stats[384:512]=shift
__global__ void bn_stats(const float* __restrict__ h, float* __restrict__ stats, int N) {
  const int F = 128;
  int f = threadIdx.x;                 // blockDim = 128
  int r0 = blockIdx.x * 256;
  int r1 = r0 + 256; if (r1 > N) r1 = N;
  float s = 0.0f, sq = 0.0f;
  for (int r = r0; r < r1; ++r) {
    float v = h[(long long)r * F + f];
    s += v; sq += v * v;
  }
  atomicAdd(&stats[f], s);
  atomicAdd(&stats[F + f], sq);
}

__global__ void bn_final(float* __restrict__ stats, const float* __restrict__ g,
                         const float* __restrict__ be, int N) {
  int f = threadIdx.x;                 // 128 threads
  float inv = 1.0f / (float)N;
  float mu = stats[f] * inv;
  float var = stats[128 + f] * inv - mu * mu;
  float sc = g[f] * rsqrtf(var + BN_EPS);
  stats[256 + f] = sc;
  stats[384 + f] = be[f] - mu * sc;
}

// h = leaky_relu(h*scale + shift) + res   (in place on h, float4-vectorized)
__global__ void bn_lrelu_res_v4(float4* __restrict__ h4, const float4* __restrict__ res4,
                                const float* __restrict__ stats, int N) {
  const int F4 = 32;                   // 128 / 4
  int idx = blockIdx.x * blockDim.x + threadIdx.x;
  if (idx >= N * F4) return;
  int f = idx % F4;
  float4 sc = ((const float4*)(stats + 256))[f];
  float4 sh = ((const float4*)(stats + 384))[f];
  float4 hv = h4[idx];
  float4 rv = res4[idx];
  float4 o;
  o.x = hv.x * sc.x + sh.x; o.x = (o.x > 0.0f ? o.x : NEG_SLOPE * o.x) + rv.x;
  o.y = hv.y * sc.y + sh.y; o.y = (o.y > 0.0f ? o.y : NEG_SLOPE * o.y) + rv.y;
  o.z = hv.z * sc.z + sh.z; o.z = (o.z > 0.0f ? o.z : NEG_SLOPE * o.z) + rv.z;
  o.w = hv.w * sc.w + sh.w; o.w = (o.w > 0.0f ? o.w : NEG_SLOPE * o.w) + rv.w;
  h4[idx] = o;
}

// ------------------------------------------------------------------ launch
static inline int cdiv(long long a, long long b) { return (int)((a + b - 1) / b); }

extern "C" void kernel_launch(void* const* d_in, const int* in_sizes, int n_in,
                              void* d_out, int out_size, void* d_ws, size_t ws_size,
                              hipStream_t stream) {
  const float*     x   = (const float*)d_in[0];
  const long long* ei  = (const long long*)d_in[1];
  const float* W1 = (const float*)d_in[2];  const float* b1 = (const float*)d_in[3];
  const float* W2 = (const float*)d_in[4];  const float* b2 = (const float*)d_in[5];
  const float* W3 = (const float*)d_in[6];  const float* b3 = (const float*)d_in[7];
  const float* g1 = (const float*)d_in[8];  const float* be1 = (const float*)d_in[9];
  const float* g2 = (const float*)d_in[10]; const float* be2 = (const float*)d_in[11];
  float* out = (float*)d_out;

  const int N = in_sizes[0] / 128;
  const int E = in_sizes[1] / 2;

  // workspace layout (floats): dis[N] | stats[512] | B0 | B1 | B2 (each N*128)
  float* ws    = (float*)d_ws;
  float* dis   = ws;
  float* stats = ws + N;
  float* B0    = stats + 512;
  float* B1    = B0 + (size_t)N * 128;
  float* B2    = B1 + (size_t)N * 128;

  const dim3 blk(256);
  const int gV4_128 = cdiv((long long)N * 32, 256);   // N*128/4 elements
  const int gV4_64  = cdiv((long long)N * 16, 256);   // N*64/4 elements
  const int gEdge   = cdiv((long long)E * 32, 256);
  const int gGemm   = cdiv(N, 128);
  const int gStats  = cdiv(N, 256);

  // degrees -> dis = (deg+1)^{-1/2}
  zero_f32<<<cdiv(N, 256), blk, 0, stream>>>(dis, N);
  deg_accum<<<cdiv(E, 256), blk, 0, stream>>>(ei, dis, E);
  deg_to_dis<<<cdiv(N, 256), blk, 0, stream>>>(dis, N);

  // ---- layer 1: h = x@W1 ; agg = scatter + selfloop + b1 ; BN+lrelu+res(x)
  gemm_wmma_f32<128><<<gGemm, blk, 0, stream>>>(x, W1, B0, N);
  agg_init_v4<<<gV4_128, blk, 0, stream>>>((const float4*)B0, dis, (const float4*)b1,
                                           (float4*)B1, N, 32);
  edge_scatter<128><<<gEdge, blk, 0, stream>>>(ei, dis, B0, B1, E);
  zero_f32<<<1, blk, 0, stream>>>(stats, 256);
  bn_stats<<<gStats, dim3(128), 0, stream>>>(B1, stats, N);
  bn_final<<<1, dim3(128), 0, stream>>>(stats, g1, be1, N);
  bn_lrelu_res_v4<<<gV4_128, blk, 0, stream>>>((float4*)B1, (const float4*)x, stats, N);

  // ---- layer 2
  gemm_wmma_f32<128><<<gGemm, blk, 0, stream>>>(B1, W2, B0, N);
  agg_init_v4<<<gV4_128, blk, 0, stream>>>((const float4*)B0, dis, (const float4*)b2,
                                           (float4*)B2, N, 32);
  edge_scatter<128><<<gEdge, blk, 0, stream>>>(ei, dis, B0, B2, E);
  zero_f32<<<1, blk, 0, stream>>>(stats, 256);
  bn_stats<<<gStats, dim3(128), 0, stream>>>(B2, stats, N);
  bn_final<<<1, dim3(128), 0, stream>>>(stats, g2, be2, N);
  bn_lrelu_res_v4<<<gV4_128, blk, 0, stream>>>((float4*)B2, (const float4*)B1, stats, N);

  // ---- output layer (F=64), aggregate straight into d_out
  gemm_wmma_f32<64><<<gGemm, blk, 0, stream>>>(B2, W3, B0, N);
  agg_init_v4<<<gV4_64, blk, 0, stream>>>((const float4*)B0, dis, (const float4*)b3,
                                          (float4*)out, N, 16);
  edge_scatter<64><<<gEdge, blk, 0, stream>>>(ei, dis, B0, out, E);
}